// Model_50371376447948
// MI455X (gfx1250) — compile-verified
//
#include <hip/hip_runtime.h>
#include <hip/hip_bf16.h>

// ---------------------------------------------------------------------------
// GNN (MetaLayer + per-graph kNN-10) forward, fp32, CDNA5 (gfx1250).
// All GEMMs run on the WMMA pipe via V_WMMA_F32_16X16X4_F32 (fp32 in/out,
// matches reference numerics; kNN selection is precision-sensitive).
// Deterministic: no float atomics; BN stats via fixed-partition two-stage
// reduction; scatter-mean via per-node edge-list collection + rank sort.
// ---------------------------------------------------------------------------

typedef float v2f __attribute__((ext_vector_type(2)));
typedef float v8f __attribute__((ext_vector_type(8)));

#define NNODE  32768           // B*M
#define BGRAPH 16
#define MNODE  2048
#define KNN    10
#define EEDGE  (NNODE * KNN)   // 327680
#define FD     256
#define INDIM  1280
#define NCLS   1000
#define BNP    512             // partial-reduction blocks for BatchNorm
#define BNEPS  1e-5f

enum { IDX_NONE = 0, IDX_ARR = 1, IDX_DIV = 2 };

// ---------------------------------------------------------------------------
// Generic WMMA GEMM:  Y[R, OC] = X[R, k0+k1] @ W[OC, k0+k1]^T (+ bias)
// X row r = concat( A0[map0(r), 0:k0],  A1[map1(r), 0:k1] )   (A1 optional)
// map(r): identity | idx[r] | r/10.  R must be a multiple of 16.
// One wave computes a 16x32 output strip (two 16x16 tiles, two accumulators):
// one A load feeds two WMMAs per K-step (halved A traffic, 2 indep chains).
// V_WMMA_F32_16X16X4_F32 lane layout (ISA 7.12.2):
//   A 16x4 : lane lm=lane&15 is row; VGPR pair holds K = (2*hi, 2*hi+1)
//   B 4x16 : lane lm is column n;    VGPR pair holds K = (2*hi, 2*hi+1)
//   D 16x16: VGPR j is row j + 8*hi, lane lm is column.
// Out-of-range columns: clamp W row to 0 and skip the store.  No zeroing is
// needed: B column n lives only in lanes n / n+16 and only affects D column n
// in those same lanes, which are exactly the lanes whose store is skipped.
// ---------------------------------------------------------------------------
__global__ void gemm_wmma_f32(const float* __restrict__ A0,
                              const int*   __restrict__ idx0, int mode0, int k0,
                              const float* __restrict__ A1,
                              const int*   __restrict__ idx1, int mode1, int k1,
                              const float* __restrict__ W,
                              const float* __restrict__ bias,
                              float* __restrict__ Y, int R, int OC) {
  const int KT     = k0 + k1;
  const int tcols  = (OC + 15) >> 4;
  const int tcols2 = (tcols + 1) >> 1;         // col-tile pairs per row tile
  const int trows  = R >> 4;
  const int wave   = blockIdx.x * (blockDim.x >> 5) + (threadIdx.x >> 5);
  if (wave >= tcols2 * trows) return;          // whole-wave exit: EXEC all-ones
  const int tr  = wave / tcols2, tcw = wave % tcols2;
  const int lane = threadIdx.x & 31;
  const int lm   = lane & 15;                  // row (A) / col (B,D) within tile
  const int hi   = lane >> 4;                  // K-pair select / +8 row offset

  const int r = tr * 16 + lm;
  long r0 = r, r1 = r;
  if (mode0 == IDX_ARR) r0 = idx0[r]; else if (mode0 == IDX_DIV) r0 = r / KNN;
  if (k1) { if (mode1 == IDX_ARR) r1 = idx1[r]; else if (mode1 == IDX_DIV) r1 = r / KNN; }
  const float* arow0 = A0 + (size_t)r0 * k0;
  const float* arow1 = k1 ? (A1 + (size_t)r1 * k1) : A0;

  const int  c0   = (tcw * 2) * 16 + lm;
  const int  c1   = c0 + 16;
  const bool cok0 = (c0 < OC);
  const bool cok1 = (c1 < OC);
  const float* wrow0 = W + (size_t)(cok0 ? c0 : 0) * KT;
  const float* wrow1 = W + (size_t)(cok1 ? c1 : 0) * KT;

  v8f acc0 = {0.f, 0.f, 0.f, 0.f, 0.f, 0.f, 0.f, 0.f};
  v8f acc1 = {0.f, 0.f, 0.f, 0.f, 0.f, 0.f, 0.f, 0.f};
  for (int kk = 0; kk < KT; kk += 4) {
    const int k = kk + 2 * hi;
    v2f a = (k < k0) ? *(const v2f*)(arow0 + k)
                     : *(const v2f*)(arow1 + (k - k0));
    v2f b0 = *(const v2f*)(wrow0 + k);
    v2f b1 = *(const v2f*)(wrow1 + k);
    acc0 = __builtin_amdgcn_wmma_f32_16x16x4_f32(false, a, false, b0,
                                                 (short)0, acc0, false, false);
    acc1 = __builtin_amdgcn_wmma_f32_16x16x4_f32(false, a, false, b1,
                                                 (short)0, acc1, false, false);
  }

  const int m0 = tr * 16 + 8 * hi;
  if (cok0) {
    const float bv = bias ? bias[c0] : 0.f;
#pragma unroll
    for (int j = 0; j < 8; ++j)
      Y[(size_t)(m0 + j) * OC + c0] = acc0[j] + bv;
  }
  if (cok1) {
    const float bv = bias ? bias[c1] : 0.f;
#pragma unroll
    for (int j = 0; j < 8; ++j)
      Y[(size_t)(m0 + j) * OC + c1] = acc1[j] + bv;
  }
}

// ---------------------------------------------------------------------------
// BatchNorm1d (training mode, biased var), deterministic two-stage reduction.
// ---------------------------------------------------------------------------
__global__ void bn_partial_k(const float* __restrict__ Y, int R,
                             float* __restrict__ part) {
  const int blk = blockIdx.x;       // 0..BNP-1, fixed row range per block
  const int c   = threadIdx.x;      // 0..255 column
  const int chunk = (R + BNP - 1) / BNP;
  const int r0 = blk * chunk;
  const int r1 = (r0 + chunk < R) ? (r0 + chunk) : R;
  float s = 0.f, ss = 0.f;
  for (int r = r0; r < r1; ++r) {
    const float v = Y[(size_t)r * FD + c];
    s += v; ss += v * v;
  }
  part[(size_t)blk * (2 * FD) + c]      = s;
  part[(size_t)blk * (2 * FD) + FD + c] = ss;
}

__global__ void bn_finalize_k(const float* __restrict__ part, int R,
                              const float* __restrict__ g,
                              const float* __restrict__ b,
                              float* __restrict__ scale,
                              float* __restrict__ shift) {
  const int c = threadIdx.x;        // 256 threads, fixed-order accumulation
  float s = 0.f, ss = 0.f;
  for (int p = 0; p < BNP; ++p) {
    s  += part[(size_t)p * (2 * FD) + c];
    ss += part[(size_t)p * (2 * FD) + FD + c];
  }
  const float inv = 1.f / (float)R;
  const float mu  = s * inv;
  float var = ss * inv - mu * mu;
  if (var < 0.f) var = 0.f;
  const float sc = g[c] * rsqrtf(var + BNEPS);
  scale[c] = sc;
  shift[c] = b[c] - mu * sc;
}

__global__ void bn_apply_k(float* __restrict__ Y, size_t total,
                           const float* __restrict__ scale,
                           const float* __restrict__ shift, int relu) {
  const size_t i = (size_t)blockIdx.x * blockDim.x + threadIdx.x;
  if (i >= total) return;
  const int c = (int)(i & (FD - 1));
  float v = Y[i] * scale[c] + shift[c];
  if (relu) v = fmaxf(v, 0.f);
  Y[i] = v;
}

// ---------------------------------------------------------------------------
// kNN graph build (per graph b of 2048 nodes, feature dim 256)
// ---------------------------------------------------------------------------
__global__ void rowsq_k(const float* __restrict__ h, float* __restrict__ sq) {
  const int i = blockIdx.x * blockDim.x + threadIdx.x;
  if (i >= NNODE) return;
  const float* r = h + (size_t)i * FD;
  float s = 0.f;
  for (int c = 0; c < FD; ++c) s += r[c] * r[c];
  sq[i] = s;
}

// D[m][n] = sq[m] + sq[n] - 2 * dot(H[m], H[n]);  diag = +inf.  WMMA Gram.
__global__ void knn_dist_k(const float* __restrict__ Hb,
                           const float* __restrict__ sqb,
                           float* __restrict__ D) {
  const int tcols = MNODE / 16;
  const int wave  = blockIdx.x * (blockDim.x >> 5) + (threadIdx.x >> 5);
  if (wave >= tcols * tcols) return;
  const int tr = wave / tcols, tc = wave % tcols;
  const int lane = threadIdx.x & 31;
  const int lm = lane & 15, hi = lane >> 4;

  const float* arow = Hb + (size_t)(tr * 16 + lm) * FD;
  const float* brow = Hb + (size_t)(tc * 16 + lm) * FD;
  v8f acc = {0.f, 0.f, 0.f, 0.f, 0.f, 0.f, 0.f, 0.f};
  for (int kk = 0; kk < FD; kk += 4) {
    v2f a = *(const v2f*)(arow + kk + 2 * hi);
    v2f b = *(const v2f*)(brow + kk + 2 * hi);
    acc = __builtin_amdgcn_wmma_f32_16x16x4_f32(false, a, false, b,
                                                (short)0, acc, false, false);
  }
  const int n  = tc * 16 + lm;
  const float sn = sqb[n];
  const int m0 = tr * 16 + 8 * hi;
#pragma unroll
  for (int j = 0; j < 8; ++j) {
    const int m = m0 + j;
    float d = sqb[m] + sn - 2.f * acc[j];
    if (m == n) d = __builtin_inff();      // loop=False
    D[(size_t)m * MNODE + n] = d;
  }
}

// Stable top-10 smallest per row (strict '<' insertion => lower index wins ties,
// matching lax.top_k on -d).  nbr[e] = global neighbor index (edge row array).
__global__ void knn_topk_k(const float* __restrict__ D, int* __restrict__ nbr,
                           int b) {
  const int i = blockIdx.x * blockDim.x + threadIdx.x;
  if (i >= MNODE) return;
  const float* row = D + (size_t)i * MNODE;
  float bv[KNN]; int bi[KNN];
#pragma unroll
  for (int t = 0; t < KNN; ++t) { bv[t] = __builtin_inff(); bi[t] = 0; }
  for (int j = 0; j < MNODE; ++j) {
    const float v = row[j];
    if (v < bv[KNN - 1]) {
      int p = KNN - 1;
      while (p > 0 && v < bv[p - 1]) { bv[p] = bv[p - 1]; bi[p] = bi[p - 1]; --p; }
      bv[p] = v; bi[p] = j;
    }
  }
  const size_t e0 = ((size_t)b * MNODE + i) * KNN;
#pragma unroll
  for (int t = 0; t < KNN; ++t) nbr[e0 + t] = b * MNODE + bi[t];
}

// ---------------------------------------------------------------------------
// Deterministic scatter-mean: agg[n] = mean over edges e with row[e]==n of t[e].
// One block per node; scan this graph's 20480-edge slice, collect matches,
// rank-sort edge ids in LDS (deterministic order), then column-wise sum.
// ---------------------------------------------------------------------------
__global__ void seg_mean_k(const float* __restrict__ t,
                           const int*   __restrict__ row,
                           float* __restrict__ agg) {
  __shared__ int list[MNODE];
  __shared__ int slist[MNODE];
  __shared__ int cnt;
  const int n  = blockIdx.x;                 // node id
  const int b  = n >> 11;                    // n / 2048
  const int e0 = b * MNODE * KNN;            // 20480 edges per graph
  if (threadIdx.x == 0) cnt = 0;
  __syncthreads();
  for (int e = e0 + threadIdx.x; e < e0 + MNODE * KNN; e += blockDim.x) {
    if (row[e] == n) { const int p = atomicAdd(&cnt, 1); list[p] = e; }
  }
  __syncthreads();
  const int L = cnt;                          // <= 2047 (in-degree bound)
  for (int j = threadIdx.x; j < L; j += blockDim.x) {   // rank sort (ids unique)
    const int v = list[j];
    int rank = 0;
    for (int q = 0; q < L; ++q) rank += (list[q] < v);
    slist[rank] = v;
  }
  __syncthreads();
  const int c = threadIdx.x;                  // 256 threads = 256 columns
  float s = 0.f;
  for (int j = 0; j < L; ++j) s += t[(size_t)slist[j] * FD + c];
  const float denom = (L > 0) ? (float)L : 1.f;
  agg[(size_t)n * FD + c] = s / denom;
}

// ---------------------------------------------------------------------------
// Host side
// ---------------------------------------------------------------------------
struct LN { const float *W, *g, *b; };

static inline void launch_gemm(const float* A0, const int* i0, int m0, int k0,
                               const float* A1, const int* i1, int m1, int k1,
                               const float* W, const float* bias,
                               float* Y, int R, int OC, hipStream_t s) {
  const int tcols2 = (((OC + 15) / 16) + 1) / 2;
  const int waves  = tcols2 * (R / 16);
  const int blocks = (waves + 7) / 8;                 // 8 waves / 256 threads
  gemm_wmma_f32<<<blocks, 256, 0, s>>>(A0, i0, m0, k0, A1, i1, m1, k1,
                                       W, bias, Y, R, OC);
}

static inline void run_bn(float* Y, int R, const float* g, const float* b,
                          int relu, float* part, float* scale, float* shift,
                          hipStream_t s) {
  bn_partial_k<<<BNP, FD, 0, s>>>(Y, R, part);
  bn_finalize_k<<<1, FD, 0, s>>>(part, R, g, b, scale, shift);
  const size_t total = (size_t)R * FD;
  bn_apply_k<<<(unsigned)((total + 255) / 256), 256, 0, s>>>(Y, total, scale,
                                                             shift, relu);
}

extern "C" void kernel_launch(void* const* d_in, const int* in_sizes, int n_in,
                              void* d_out, int out_size, void* d_ws, size_t ws_size,
                              hipStream_t stream) {
  // ---- decode inputs (setup_inputs() insertion order, DFS through dicts) ----
  int i = 0;
  const float* x = (const float*)d_in[i++];             // [N, 1280]
  auto getLN = [&]() {
    LN p; p.W = (const float*)d_in[i++]; p.g = (const float*)d_in[i++];
    p.b = (const float*)d_in[i++]; return p;
  };
  LN project = getLN();
  LN L[3][6];                                           // e1,e2,n1a,n1b,n2a,n2b
  for (int l = 0; l < 3; ++l)
    for (int j = 0; j < 6; ++j) L[l][j] = getLN();
  const float* outW = (const float*)d_in[i++];          // [1000, 256]
  const float* outb = (const float*)d_in[i++];          // [1000]

  // ---- workspace carve-out (aligned to 256B) ----
  size_t off = 0;
  auto alloc = [&](size_t bytes) {
    size_t o = off; off += (bytes + 255) & ~(size_t)255; return o;
  };
  char* ws = (char*)d_ws;
  float* hA   = (float*)(ws + alloc((size_t)NNODE * FD * 4));
  float* hB   = (float*)(ws + alloc((size_t)NNODE * FD * 4));
  float* t1   = (float*)(ws + alloc((size_t)EEDGE * FD * 4));
  float* t2   = (float*)(ws + alloc((size_t)EEDGE * FD * 4));
  float* t3   = (float*)(ws + alloc((size_t)NNODE * FD * 4));
  float* agg  = (float*)(ws + alloc((size_t)NNODE * FD * 4));
  int*   nbr  = (int*)  (ws + alloc((size_t)EEDGE * 4));
  float* sq   = (float*)(ws + alloc((size_t)NNODE * 4));
  float* Dbuf = (float*)(ws + alloc((size_t)MNODE * MNODE * 4));
  float* part = (float*)(ws + alloc((size_t)BNP * 2 * FD * 4));
  float* bnsc = (float*)(ws + alloc((size_t)FD * 4));
  float* bnsh = (float*)(ws + alloc((size_t)FD * 4));

  float* out = (float*)d_out;                           // [N, 1000]

  // ---- project: h = BN(x @ Wp^T) ----
  launch_gemm(x, nullptr, IDX_NONE, INDIM, nullptr, nullptr, IDX_NONE, 0,
              project.W, nullptr, hA, NNODE, FD, stream);
  run_bn(hA, NNODE, project.g, project.b, 0, part, bnsc, bnsh, stream);

  float* hcur = hA;
  float* hnxt = hB;

  for (int l = 0; l < 3; ++l) {
    LN e1 = L[l][0], e2 = L[l][1], n1a = L[l][2], n1b = L[l][3],
       n2a = L[l][4], n2b = L[l][5];

    // ---- kNN graph on hcur (per graph: Gram via WMMA, then stable top-10) ----
    rowsq_k<<<NNODE / 256, 256, 0, stream>>>(hcur, sq);
    for (int b = 0; b < BGRAPH; ++b) {
      const int waves  = (MNODE / 16) * (MNODE / 16);
      knn_dist_k<<<(waves + 7) / 8, 256, 0, stream>>>(
          hcur + (size_t)b * MNODE * FD, sq + (size_t)b * MNODE, Dbuf);
      knn_topk_k<<<MNODE / 256, 256, 0, stream>>>(Dbuf, nbr, b);
    }

    // ---- EdgeModel: e = BN(relu(BN(cat(h[row], h[col]) @ We1^T)) @ We2^T) ----
    launch_gemm(hcur, nbr, IDX_ARR, FD, hcur, nullptr, IDX_DIV, FD,
                e1.W, nullptr, t1, EEDGE, FD, stream);
    run_bn(t1, EEDGE, e1.g, e1.b, 1, part, bnsc, bnsh, stream);
    launch_gemm(t1, nullptr, IDX_NONE, FD, nullptr, nullptr, IDX_NONE, 0,
                e2.W, nullptr, t2, EEDGE, FD, stream);
    run_bn(t2, EEDGE, e2.g, e2.b, 0, part, bnsc, bnsh, stream);

    // ---- NodeModel part 1: h1 = BN(relu(BN(cat(h[col], e) @ Wn1a^T)) @ Wn1b^T)
    launch_gemm(hcur, nullptr, IDX_DIV, FD, t2, nullptr, IDX_NONE, FD,
                n1a.W, nullptr, t1, EEDGE, FD, stream);
    run_bn(t1, EEDGE, n1a.g, n1a.b, 1, part, bnsc, bnsh, stream);
    launch_gemm(t1, nullptr, IDX_NONE, FD, nullptr, nullptr, IDX_NONE, 0,
                n1b.W, nullptr, t2, EEDGE, FD, stream);
    run_bn(t2, EEDGE, n1b.g, n1b.b, 0, part, bnsc, bnsh, stream);

    // ---- scatter-mean onto row ----
    seg_mean_k<<<NNODE, FD, 0, stream>>>(t2, nbr, agg);

    // ---- NodeModel part 2: h' = BN(relu(BN(cat(h, agg) @ Wn2a^T)) @ Wn2b^T) --
    launch_gemm(hcur, nullptr, IDX_NONE, FD, agg, nullptr, IDX_NONE, FD,
                n2a.W, nullptr, t3, NNODE, FD, stream);
    run_bn(t3, NNODE, n2a.g, n2a.b, 1, part, bnsc, bnsh, stream);
    launch_gemm(t3, nullptr, IDX_NONE, FD, nullptr, nullptr, IDX_NONE, 0,
                n2b.W, nullptr, hnxt, NNODE, FD, stream);
    run_bn(hnxt, NNODE, n2b.g, n2b.b, 0, part, bnsc, bnsh, stream);

    float* tmp = hcur; hcur = hnxt; hnxt = tmp;
  }

  // ---- classifier: out = h @ out_W^T + out_b ----
  launch_gemm(hcur, nullptr, IDX_NONE, FD, nullptr, nullptr, IDX_NONE, 0,
              outW, outb, out, NNODE, NCLS, stream);
}